// KANLayer_89026082112115
// MI455X (gfx1250) — compile-verified
//
#include <hip/hip_runtime.h>
#include <cmath>

// KAN layer, fully fused for gfx1250 (MI455X, wave32).
// out[o,n] = sum_i [ w_b[i,o]*silu(x[i,n]) + w_s[i,o]*sum_c cp[i,o,c]*B[i,n,c] ]
// B is a degree-3 B-spline basis: only 4 consecutive nonzeros per (i,n).
// Strategy: per block, expand the 4-sparse basis of 2 i-values into a dense
// padded LDS tile (rows 0..64 = basis, row 68 = silu(x), other rows zero) and
// drive both reductions through V_WMMA_F32_16X16X4_F32 rank-4 updates.

#define IN_DIM  128
#define OUT_DIM 128
#define NPTS    8192
#define NCTRL   65
#define NKNOT   69
#define NTILE   64      // n columns per block
#define CSTRIDE 73      // padded c stride: 73 % 64 = 9 (odd) -> conflict-free lane pattern
#define SLOTS   2       // i values expanded per batch
#define THREADS 256

typedef __attribute__((ext_vector_type(2))) float v2f;
typedef __attribute__((ext_vector_type(8))) float v8f;

static __device__ __forceinline__ v8f wmma_k4(v2f a, v2f b, v8f c) {
  // D(16o x 16n) += A(16o x 4c) * B(4c x 16n), f32 accumulate
  return __builtin_amdgcn_wmma_f32_16x16x4_f32(false, a, false, b, (short)0, c,
                                               false, false);
}

__global__ __launch_bounds__(THREADS) void kan_fused(
    const float* __restrict__ x,      // (128, 8192)
    const float* __restrict__ w_b,    // (128, 128) [i][o]
    const float* __restrict__ w_s,    // (128, 128) [i][o]
    const float* __restrict__ gp,     // (69) knots
    const float* __restrict__ cp,     // (128, 128, 65) [i][o][c]
    float* __restrict__ out)          // (128, 8192) [o][n]
{
  __shared__ float lds[SLOTS * NTILE * CSTRIDE + 72];
  float* kn = &lds[SLOTS * NTILE * CSTRIDE];

  const int t  = threadIdx.x;
  const int n0 = blockIdx.x * NTILE;

  // ---- init: zero dense tiles (incl. permanently-zero pad rows), stage knots
  for (int idx = t; idx < SLOTS * NTILE * CSTRIDE + 72; idx += THREADS)
    lds[idx] = 0.f;
  __syncthreads();
  if (t < NKNOT) kn[t] = gp[t];
  __syncthreads();

  // ---- scatter identity: 128 producer threads, (slot, n) = (t/64, t%64)
  const bool producer = (t < SLOTS * NTILE);
  const int  slot = t >> 6;
  const int  nn   = t & 63;
  int prev_m = 3;

  // Loop-invariant knot scale, from GLOBAL (not LDS) so it provably can't
  // alias the scatter stores and stays hoisted: one load pair, one divide.
  const float t0k  = gp[0];
  const float invh = 68.f / (gp[NKNOT - 1] - t0k);

  // ---- wmma identity: 8 waves, each owns 2 o-tiles x 2 n-tiles
  const int lane  = t & 31;
  const int wave  = t >> 5;
  const int hi    = lane >> 4;         // 0: K cols 0/1, 1: K cols 2/3
  const int lo    = lane & 15;
  const int obase = (wave & 3) * 32;   // 4 o-pairs cover o = 0..127
  const int nbase = (wave >> 2) * 32;  // 2 n-pairs cover n-tile 0..63

  v8f acc[2][2] = {};

  const int hi2 = 2 * hi;

  #pragma unroll 1
  for (int batch = 0; batch < IN_DIM / SLOTS; ++batch) {
    const int i0 = batch * SLOTS;

    // ======== compute phase (VGPR only; overlaps prior batch's WMMA) ========
    float sv = 0.f, Nb[4] = {1.f, 0.f, 0.f, 0.f};
    int m = 3;
    if (producer) {
      const int ip = i0 + slot;
      const float xv = x[(size_t)ip * NPTS + n0 + nn];
      sv = xv * (1.f / (1.f + __expf(-xv)));             // silu (fast exp)

      // knot interval: uniform estimate + branchless FP fix-up
      m = (int)floorf((xv - t0k) * invh);
      m = m < 3 ? 3 : (m > 64 ? 64 : m);
      {
        const float km  = kn[m];
        const float km1 = kn[m + 1];
        const int dec = (xv < km  && m > 3)  ? 1 : 0;
        const int inc = (!dec && xv >= km1 && m < 64) ? 1 : 0;
        m = m - dec + inc;
      }

      // local de Boor (degree 3): Nb[k] = B_{m-3+k}(xv)
      float left[4], right[4];
      #pragma unroll
      for (int j = 1; j <= 3; ++j) {
        left[j]  = xv - kn[m + 1 - j];
        right[j] = kn[m + j] - xv;
        float saved = 0.f;
        #pragma unroll
        for (int r = 0; r < j; ++r) {
          const float temp = __fdividef(Nb[r], right[r + 1] + left[j - r]);
          Nb[r] = saved + right[r + 1] * temp;
          saved = left[j - r] * temp;
        }
        Nb[j] = saved;
      }
    }

    __syncthreads();   // previous batch's WMMA reads are done

    // ======== scatter: clear old band, write new band + silu row ========
    if (producer) {
      float* drow = &lds[(slot * NTILE + nn) * CSTRIDE];
      drow[prev_m - 3] = 0.f; drow[prev_m - 2] = 0.f;
      drow[prev_m - 1] = 0.f; drow[prev_m]     = 0.f;
      drow[m - 3] = Nb[0]; drow[m - 2] = Nb[1];
      drow[m - 1] = Nb[2]; drow[m]     = Nb[3];
      drow[68]    = sv;
      prev_m = m;
    }
    __syncthreads();

    // ======== WMMA phase: K = 18 chunks x 4 (65 ctrl + pad + silu + pad) ====
    #pragma unroll
    for (int s2 = 0; s2 < SLOTS; ++s2) {
      const int iw = i0 + s2;
      const int o0 = obase + lo;            // A rows for o-tile 0
      const int o1 = obase + 16 + lo;       // A rows for o-tile 1
      const float ws0 = w_s[iw * OUT_DIM + o0];
      const float ws1 = w_s[iw * OUT_DIM + o1];
      const float* cp0 = cp + ((size_t)iw * OUT_DIM + o0) * NCTRL;
      const float* cp1 = cp + ((size_t)iw * OUT_DIM + o1) * NCTRL;
      const float* br0 = &lds[(s2 * NTILE + nbase + lo) * CSTRIDE];
      const float* br1 = &lds[(s2 * NTILE + nbase + 16 + lo) * CSTRIDE];

      if (iw + SLOTS < IN_DIM) {            // pull next batch's cp rows
        __builtin_prefetch(cp0 + (size_t)SLOTS * OUT_DIM * NCTRL, 0, 1);
        __builtin_prefetch(cp1 + (size_t)SLOTS * OUT_DIM * NCTRL, 0, 1);
      }

      #pragma unroll
      for (int cc = 0; cc < 18; ++cc) {
        const int c0 = 4 * cc + hi2;        // this lane's K columns c0, c0+1
        v2f a0, a1, b0, b1;
        if (cc < 16) {                      // spline body: w_s * cp
          a0.x = ws0 * cp0[c0]; a0.y = ws0 * cp0[c0 + 1];
          a1.x = ws1 * cp1[c0]; a1.y = ws1 * cp1[c0 + 1];
        } else if (cc == 16) {              // tail: only c=64 valid (stay in bounds)
          a0.x = hi ? 0.f : ws0 * cp0[64];  a0.y = 0.f;
          a1.x = hi ? 0.f : ws1 * cp1[64];  a1.y = 0.f;
        } else {                            // silu chunk: w_b against row 68
          a0.x = hi ? 0.f : w_b[iw * OUT_DIM + o0];  a0.y = 0.f;
          a1.x = hi ? 0.f : w_b[iw * OUT_DIM + o1];  a1.y = 0.f;
        }
        b0.x = br0[c0]; b0.y = br0[c0 + 1]; // LDS dense basis rows
        b1.x = br1[c0]; b1.y = br1[c0 + 1];

        acc[0][0] = wmma_k4(a0, b0, acc[0][0]);
        acc[0][1] = wmma_k4(a0, b1, acc[0][1]);
        acc[1][0] = wmma_k4(a1, b0, acc[1][0]);
        acc[1][1] = wmma_k4(a1, b1, acc[1][1]);
      }
    }
  }

  // ======== epilogue: C/D layout -> out[o][n] ========
  #pragma unroll
  for (int ot = 0; ot < 2; ++ot) {
    #pragma unroll
    for (int nt = 0; nt < 2; ++nt) {
      const v8f a = acc[ot][nt];
      const int nidx = n0 + nbase + nt * 16 + lo;
      #pragma unroll
      for (int r = 0; r < 8; ++r) {
        const int o = obase + ot * 16 + hi * 8 + r;  // lanes 16-31 hold M=8..15
        out[(size_t)o * NPTS + nidx] = a[r];
      }
    }
  }
}

extern "C" void kernel_launch(void* const* d_in, const int* in_sizes, int n_in,
                              void* d_out, int out_size, void* d_ws, size_t ws_size,
                              hipStream_t stream) {
  (void)in_sizes; (void)n_in; (void)out_size; (void)d_ws; (void)ws_size;
  const float* x   = (const float*)d_in[0];
  const float* w_b = (const float*)d_in[1];
  const float* w_s = (const float*)d_in[2];
  const float* gp  = (const float*)d_in[3];
  const float* cp  = (const float*)d_in[4];
  float* out = (float*)d_out;

  kan_fused<<<dim3(NPTS / NTILE), dim3(THREADS), 0, stream>>>(
      x, w_b, w_s, gp, cp, out);
}